// LingUNet_11622181503291
// MI455X (gfx1250) — compile-verified
//
#include <hip/hip_runtime.h>
#include <math.h>

typedef __attribute__((ext_vector_type(2))) float fp32x2;
typedef __attribute__((ext_vector_type(8))) float fp32x8;

static __device__ __forceinline__ float sigmoidf_(float x) {
  return 1.0f / (1.0f + expf(-x));
}

// ---------------------------------------------------------------------------
// Generic batched GEMM via V_WMMA_F32_16X16X4_F32.
// C[M,N] = A[M,K] @ B[K,N] (+bias, optional relu), row major.
// REQUIRES: M % 16 == 0, K % 4 == 0 (call sites pad to satisfy this).
// Workgroup = 256 threads = 8 waves; tile = 16 rows x 128 cols (wave j -> 16 cols).
// Fragment layouts per CDNA5 ISA 7.12.2:
//   A 16x4 : lane l -> m = l&15, k = (l>>4)*2 + vgpr
//   B 4x16 : lane l -> n = l&15, k = (l>>4)*2 + vgpr
//   D 16x16: lane l, vgpr v -> m = v + 8*(l>>4), n = l&15
// Inner loop: pointer-bump addressing + 1-deep software pipeline so the
// next fragments' global loads are in flight during the current wmma.
// ---------------------------------------------------------------------------
__global__ void __launch_bounds__(256)
gemm_wmma_f32(const float* __restrict__ A, const float* __restrict__ B,
              float* __restrict__ C, int M, int N, int K,
              int lda, int ldb, int ldc,
              long sA, long sB, long sC,
              const float* __restrict__ bias, int relu)
{
  int lane = threadIdx.x & 31;
  int wave = threadIdx.x >> 5;
  int n0 = blockIdx.x * 128 + wave * 16;
  if (n0 >= N) return;                      // wave-uniform exit
  int m0 = blockIdx.y * 16;
  const float* Ab = A + (long)blockIdx.z * sA;
  const float* Bb = B + (long)blockIdx.z * sB;
  float* Cb = C + (long)blockIdx.z * sC;

  int koff = (lane >> 4) * 2;
  int ncol = n0 + (lane & 15);
  int mrow = m0 + (lane & 15);

  const float* ap  = Ab + (long)mrow * lda + koff;
  const float* bp0 = Bb + (long)koff * ldb + ncol;
  const float* bp1 = bp0 + ldb;
  const long bstep = 4L * ldb;

  fp32x8 acc = {};
  fp32x2 a, bf;
  a.x  = ap[0];  a.y  = ap[1];
  bf.x = bp0[0]; bf.y = bp1[0];

#pragma unroll 2
  for (int k = 4; k < K; k += 4) {
    ap += 4; bp0 += bstep; bp1 += bstep;
    fp32x2 an, bn;
    an.x = ap[0];  an.y = ap[1];       // prefetch next fragments
    bn.x = bp0[0]; bn.y = bp1[0];
    acc = __builtin_amdgcn_wmma_f32_16x16x4_f32(false, a, false, bf,
                                                (short)0, acc, false, false);
    a = an; bf = bn;
  }
  acc = __builtin_amdgcn_wmma_f32_16x16x4_f32(false, a, false, bf,
                                              (short)0, acc, false, false);

  float bv = bias ? bias[ncol] : 0.0f;
  int mbase = (lane >> 4) * 8;
  float* crow = Cb + (long)(m0 + mbase) * ldc + ncol;
#pragma unroll
  for (int v = 0; v < 8; ++v) {
    float val = acc[v] + bv;
    if (relu) val = fmaxf(val, 0.0f);
    crow[(long)v * ldc] = val;
  }
}

// ---------------------------------------------------------------------------
// 5x5 SAME conv (== this model's conv_transpose at stride 1), NHWC, Cout=128,
// implicit GEMM with WMMA f32. Workgroup: 16 pixels (along W) x 128 out-chans,
// 8 waves. Input channels chunked by 64 through LDS (padded stride 68 floats
// -> conflict-free b64 A-fragment reads). inB (optional) supplies channels
// 128..255 for the concat([h2,g1]) layer without materializing the concat.
// ---------------------------------------------------------------------------
#define CONV_CHUNK 64
#define TSTR 68

__global__ void __launch_bounds__(256)
conv5x5_wmma(const float* __restrict__ inA, const float* __restrict__ inB,
             const float* __restrict__ wgt, const float* __restrict__ bias,
             float* __restrict__ out, int relu)
{
  const int H = 100, W = 464, CO = 128;
  __shared__ float tile[5 * 20 * TSTR];

  int lane = threadIdx.x & 31;
  int wave = threadIdx.x >> 5;
  int w0 = blockIdx.x * 16;
  int h  = blockIdx.y;
  int b  = blockIdx.z;
  int cinTot = inB ? 256 : 128;
  long plane = (long)H * W * 128;
  const float* baseA = inA + (long)b * plane;
  const float* baseB = inB ? (inB + (long)b * plane) : (const float*)0;

  int ncol = wave * 16 + (lane & 15);
  int koff = (lane >> 4) * 2;
  int m    = lane & 15;

  fp32x8 acc = {};
  for (int c0 = 0; c0 < cinTot; c0 += CONV_CHUNK) {
    const float* src = (c0 < 128) ? baseA : baseB;
    int cb = c0 & 127;
    __syncthreads();
    // cooperative fill: 5 rows x 20 cols x 64 channels (float4), zero pad OOB
    for (int idx = threadIdx.x; idx < 5 * 20 * (CONV_CHUNK / 4); idx += 256) {
      int c4 = idx & (CONV_CHUNK / 4 - 1);        // CONV_CHUNK/4 == 16
      int x  = (idx >> 4) % 20;
      int r  = (idx >> 4) / 20;
      int hh = h - 2 + r, ww = w0 - 2 + x;
      float4 v = {0.0f, 0.0f, 0.0f, 0.0f};
      if (hh >= 0 && hh < H && ww >= 0 && ww < W)
        v = *(const float4*)(src + ((long)hh * W + ww) * 128 + cb + c4 * 4);
      *(float4*)(&tile[(r * 20 + x) * TSTR + c4 * 4]) = v;
    }
    __syncthreads();

    for (int dy = 0; dy < 5; ++dy) {
      for (int dx = 0; dx < 5; ++dx) {
        const float* tptr = &tile[(dy * 20 + m + dx) * TSTR + koff];
        const float* wrow = wgt + ((long)((dy * 5 + dx) * cinTot + c0) + koff) * CO + ncol;
#pragma unroll
        for (int kk = 0; kk < CONV_CHUNK; kk += 4) {
          fp32x2 a, bf;
          a.x = tptr[kk];
          a.y = tptr[kk + 1];
          bf.x = wrow[(long)kk * CO];
          bf.y = wrow[(long)(kk + 1) * CO];
          acc = __builtin_amdgcn_wmma_f32_16x16x4_f32(false, a, false, bf,
                                                      (short)0, acc, false, false);
        }
      }
    }
  }

  float bv = bias[ncol];
  int mbase = (lane >> 4) * 8;
#pragma unroll
  for (int v = 0; v < 8; ++v) {
    int w = w0 + mbase + v;   // W=464 divisible by 16 -> always in range
    float val = acc[v] + bv;
    if (relu) val = fmaxf(val, 0.0f);
    out[(((long)b * H + h) * W + w) * 128 + ncol] = val;
  }
}

// ---------------------------------------------------------------------------
// Embedding gather: forward text and length-reversed text (bwd LSTM input).
// ---------------------------------------------------------------------------
__global__ void embed_gather(const float* __restrict__ table,
                             const int* __restrict__ ids,
                             const int* __restrict__ lens,
                             float* __restrict__ tF, float* __restrict__ tB)
{
  const int B = 4, T = 128, E = 300;
  int idx = blockIdx.x * blockDim.x + threadIdx.x;
  if (idx >= B * T * E) return;
  int e = idx % E;
  int t = (idx / E) % T;
  int b = idx / (E * T);
  int L = lens[b];
  int tb = (t < L) ? (L - 1 - t) : t;
  tF[idx] = table[(long)ids[b * T + t] * E + e];
  tB[idx] = table[(long)ids[b * T + tb] * E + e];
}

// ---------------------------------------------------------------------------
// LSTM recurrence. xg = x@Wx + b is precomputed (WMMA GEMM). One workgroup per
// (batch, direction); h/c/gates live in LDS. Accumulates the masked (t < L)
// h-sum directly -> text_embed = acc/T, so (B,T,1200) is never materialized.
// (masked sum of re-reversed bwd outputs == masked sum of bwd scan outputs)
// ---------------------------------------------------------------------------
__global__ void __launch_bounds__(1024)
lstm_scan(const float* __restrict__ xgF, const float* __restrict__ xgB,
          const float* __restrict__ WhF, const float* __restrict__ WhB,
          const int* __restrict__ lens, float* __restrict__ tembed)
{
  const int T = 128, HID = 600, G4 = 2400;
  __shared__ float hbuf[600], cbuf[600], abuf[600], gbuf[2400];
  int b = blockIdx.x, dir = blockIdx.y;
  const float* xg = (dir ? xgB : xgF) + (long)b * T * G4;
  const float* Wh = dir ? WhB : WhF;
  int L = lens[b];
  int tid = threadIdx.x, bs = blockDim.x;

  for (int i = tid; i < HID; i += bs) { hbuf[i] = 0.f; cbuf[i] = 0.f; abuf[i] = 0.f; }
  __syncthreads();

  for (int t = 0; t < T; ++t) {
    const float* xgt = xg + (long)t * G4;
    for (int j = tid; j < G4; j += bs) {
      float s = 0.f;
      const float* wp = Wh + j;
#pragma unroll 4
      for (int u = 0; u < HID; ++u) s += hbuf[u] * wp[(long)u * G4];
      gbuf[j] = xgt[j] + s;
    }
    __syncthreads();
    for (int u = tid; u < HID; u += bs) {
      float gi = gbuf[u], gf = gbuf[HID + u], gc = gbuf[2 * HID + u], go = gbuf[3 * HID + u];
      float cn = sigmoidf_(gf) * cbuf[u] + sigmoidf_(gi) * tanhf(gc);
      float hn = sigmoidf_(go) * tanhf(cn);
      cbuf[u] = cn; hbuf[u] = hn;
      if (t < L) abuf[u] += hn;
    }
    __syncthreads();
  }
  for (int u = tid; u < HID; u += bs)
    tembed[b * 1200 + dir * 600 + u] = abuf[u] * (1.0f / (float)T);
}

// ---------------------------------------------------------------------------
// Final projection: logits[p] = dot(o2[p,:], out_W). One wave per pixel.
// ---------------------------------------------------------------------------
__global__ void __launch_bounds__(256)
out_proj(const float* __restrict__ x, const float* __restrict__ w,
         float* __restrict__ y, int n)
{
  int gw = (int)((blockIdx.x * blockDim.x + threadIdx.x) >> 5);
  int lane = threadIdx.x & 31;
  if (gw >= n) return;
  const float* row = x + (long)gw * 128;
  float s = 0.f;
  for (int k = lane; k < 128; k += 32) s += row[k] * w[k];
  for (int o = 16; o > 0; o >>= 1) s += __shfl_xor(s, o, 32);
  if (lane == 0) y[gw] = s;
}

// ---------------------------------------------------------------------------
// Row softmax over N=46400, one workgroup (1024 threads) per batch row.
// ---------------------------------------------------------------------------
__global__ void __launch_bounds__(1024)
softmax_rows(const float* __restrict__ x, float* __restrict__ y, int N)
{
  __shared__ float wred[32];
  __shared__ float sval;
  int b = blockIdx.x;
  const float* row = x + (long)b * N;
  float* orow = y + (long)b * N;
  int tid = threadIdx.x, lane = tid & 31, wid = tid >> 5;
  int nw = blockDim.x >> 5;

  float m = -3.4e38f;
  for (int i = tid; i < N; i += blockDim.x) m = fmaxf(m, row[i]);
  for (int o = 16; o > 0; o >>= 1) m = fmaxf(m, __shfl_xor(m, o, 32));
  if (lane == 0) wred[wid] = m;
  __syncthreads();
  if (wid == 0) {
    float v = (lane < nw) ? wred[lane] : -3.4e38f;
    for (int o = 16; o > 0; o >>= 1) v = fmaxf(v, __shfl_xor(v, o, 32));
    if (lane == 0) sval = v;
  }
  __syncthreads();
  float mx = sval;

  float s = 0.f;
  for (int i = tid; i < N; i += blockDim.x) s += expf(row[i] - mx);
  for (int o = 16; o > 0; o >>= 1) s += __shfl_xor(s, o, 32);
  __syncthreads();
  if (lane == 0) wred[wid] = s;
  __syncthreads();
  if (wid == 0) {
    float v = (lane < nw) ? wred[lane] : 0.f;
    for (int o = 16; o > 0; o >>= 1) v += __shfl_xor(v, o, 32);
    if (lane == 0) sval = v;
  }
  __syncthreads();
  float inv = 1.0f / sval;
  for (int i = tid; i < N; i += blockDim.x) orow[i] = expf(row[i] - mx) * inv;
}

// ---------------------------------------------------------------------------
static void launch_gemm(const float* A, const float* B, float* C,
                        int M, int N, int K, int lda, int ldb, int ldc,
                        long sA, long sB, long sC, int batch,
                        const float* bias, int relu, hipStream_t s)
{
  dim3 grid((N + 127) / 128, (M + 15) / 16, batch);
  gemm_wmma_f32<<<grid, 256, 0, s>>>(A, B, C, M, N, K, lda, ldb, ldc,
                                     sA, sB, sC, bias, relu);
}

extern "C" void kernel_launch(void* const* d_in, const int* in_sizes, int n_in,
                              void* d_out, int out_size, void* d_ws, size_t ws_size,
                              hipStream_t stream)
{
  const float* image  = (const float*)d_in[0];
  const int*   instr  = (const int*)d_in[1];
  const int*   lens   = (const int*)d_in[2];
  const float* table  = (const float*)d_in[3];
  const float* fwd_Wx = (const float*)d_in[4];
  const float* fwd_Wh = (const float*)d_in[5];
  const float* fwd_b  = (const float*)d_in[6];
  const float* bwd_Wx = (const float*)d_in[7];
  const float* bwd_Wh = (const float*)d_in[8];
  const float* bwd_b  = (const float*)d_in[9];
  const float* k1W = (const float*)d_in[10];
  const float* k1b = (const float*)d_in[11];
  const float* k2W = (const float*)d_in[12];
  const float* k2b = (const float*)d_in[13];
  const float* c1w = (const float*)d_in[14];
  const float* c1b = (const float*)d_in[15];
  const float* c2w = (const float*)d_in[16];
  const float* c2b = (const float*)d_in[17];
  const float* dc1w = (const float*)d_in[18];
  const float* dc1b = (const float*)d_in[19];
  const float* dc2w = (const float*)d_in[20];
  const float* dc2b = (const float*)d_in[21];
  const float* d1W = (const float*)d_in[22];
  const float* d1b = (const float*)d_in[23];
  const float* d2W = (const float*)d_in[24];
  const float* d2b = (const float*)d_in[25];
  const float* outW = (const float*)d_in[26];

  const long HW  = 100L * 464;      // 46400
  const long PIX = 4L * HW;         // 185600
  const long SLOT = PIX * 128;      // 23,756,800 floats

  float* base = (float*)d_ws;
  long off = 0;
  auto take = [&](long n) { float* p = base + off; off += (n + 63) & ~63L; return p; };
  float* slotA  = take(SLOT);           // f1 -> h2 (see schedule)
  float* slotB  = take(SLOT);           // f2 -> h1
  float* slotC  = take(SLOT);           // g1 -> o2
  float* slotD  = take(SLOT);           // g2 -> o1
  float* textF  = take(512L * 300);
  float* textB  = take(512L * 300);
  float* xgF    = take(512L * 2400);
  float* xgB    = take(512L * 2400);
  float* temb   = take(16L * 1200);     // rows 0..3 real, 4..15 = M padding
  float* kern1  = take(16L * 16384);    // rows 0..3 real (4 batches of 128x128)
  float* kern2  = take(16L * 16384);
  float* logits = take(PIX);
  (void)ws_size; (void)in_sizes; (void)n_in; (void)out_size;

  // --- text pipeline -------------------------------------------------------
  embed_gather<<<(4 * 128 * 300 + 255) / 256, 256, 0, stream>>>(table, instr, lens, textF, textB);
  // xg = text @ Wx + b   (M=512, K=300, N=2400)
  launch_gemm(textF, fwd_Wx, xgF, 512, 2400, 300, 300, 2400, 2400, 0, 0, 0, 1, fwd_b, 0, stream);
  launch_gemm(textB, bwd_Wx, xgB, 512, 2400, 300, 300, 2400, 2400, 0, 0, 0, 1, bwd_b, 0, stream);
  lstm_scan<<<dim3(4, 2), 1024, 0, stream>>>(xgF, xgB, fwd_Wh, bwd_Wh, lens, temb);
  // kern = te @ k_W + k_b  (M padded 4->16; rows 4..15 of kern are dead)
  launch_gemm(temb,       k1W, kern1, 16, 16384, 600, 1200, 16384, 16384, 0, 0, 0, 1, k1b, 0, stream);
  launch_gemm(temb + 600, k2W, kern2, 16, 16384, 600, 1200, 16384, 16384, 0, 0, 0, 1, k2b, 0, stream);

  // --- image pipeline ------------------------------------------------------
  dim3 cgrid(464 / 16, 100, 4);
  // f1 = conv5(image) -> A ; f2 = conv5(f1) -> B
  conv5x5_wmma<<<cgrid, 256, 0, stream>>>(image, nullptr, c1w, c1b, slotA, 1);
  conv5x5_wmma<<<cgrid, 256, 0, stream>>>(slotA, nullptr, c2w, c2b, slotB, 1);
  // g1 = f1 @ kern1[b] -> C ; g2 = f2 @ kern2[b] -> D   (per-batch 128x128)
  launch_gemm(slotA, kern1, slotC, (int)HW, 128, 128, 128, 128, 128,
              HW * 128, 128L * 128, HW * 128, 4, nullptr, 0, stream);
  launch_gemm(slotB, kern2, slotD, (int)HW, 128, 128, 128, 128, 128,
              HW * 128, 128L * 128, HW * 128, 4, nullptr, 0, stream);
  // h2 = deconv5(g2) -> A (f1 dead) ; h1 = deconv5(concat(h2,g1)) -> B (f2 dead)
  conv5x5_wmma<<<cgrid, 256, 0, stream>>>(slotD, nullptr, dc2w, dc2b, slotA, 1);
  conv5x5_wmma<<<cgrid, 256, 0, stream>>>(slotA, slotC, dc1w, dc1b, slotB, 1);
  // o1 = relu(h1 @ d1_W + b) -> D (g2 dead) ; o2 = relu(o1 @ d2_W + b) -> C (g1 dead)
  launch_gemm(slotB, d1W, slotD, (int)PIX, 128, 128, 128, 128, 128, 0, 0, 0, 1, d1b, 1, stream);
  launch_gemm(slotD, d2W, slotC, (int)PIX, 128, 128, 128, 128, 128, 0, 0, 0, 1, d2b, 1, stream);
  // logits + softmax
  out_proj<<<(unsigned)((PIX * 32 + 255) / 256), 256, 0, stream>>>(slotC, outW, logits, (int)PIX);
  softmax_rows<<<4, 1024, 0, stream>>>(logits, (float*)d_out, (int)HW);
}